// SPDLayer_68015102099637
// MI455X (gfx1250) — compile-verified
//
#include <hip/hip_runtime.h>
#include <hip/hip_bf16.h>
#include <cstdint>

// ---------------------------------------------------------------------------
// SPD layer: x(B,6) -> L(3x3 lower-tri, softplus diag, entries 3,4 zeroed)
//            out = L @ L^T  (B,3,3)
// With x3=x4=0:   out = [[d0^2,  d0*x1,      0],
//                        [d0*x1, x1^2+d2^2,  0],
//                        [0,     0,          d5^2]]   di = softplus(xi)+1e-12
//
// Purely HBM-bound (60 B/sample, ~0.3 flop/B; 503 MB total -> ~21.6 us floor
// at 23.3 TB/s). Data movement via the CDNA5 Tensor Data Mover (TENSORcnt)
// with NT cache hints (working set >> 192 MB L2, zero reuse); softplus done
// with hw v_exp/v_log + a correction term so VALU stays well under DMA time.
// ---------------------------------------------------------------------------

typedef unsigned int u32x4 __attribute__((ext_vector_type(4)));
typedef int          i32x4 __attribute__((ext_vector_type(4)));
typedef int          i32x8 __attribute__((ext_vector_type(8)));

#define THREADS 256
#define SPT     4                     // samples per thread
#define TILE    (THREADS * SPT)       // 1024 samples per workgroup
#define IN_DW   (TILE * 6)            // 6144 dwords = 24 KB
#define OUT_DW  (TILE * 9)            // 9216 dwords = 36 KB

// gfx12-style cpol: TH[2:0], scope[4:3]
#define CPOL_LOAD_NT  0x1             // TH=NT, scope=WGP
#define CPOL_STORE_NT 0x11            // TH=NT, scope=DEV (rinse stream to HBM)

// ---- Tensor DMA descriptor builders (CDNA5 ISA ch.8) ----------------------
// Group 0: [1:0]=count=1, [63:32]=lds_addr, [120:64]=global_addr, [127:126]=type=2
__device__ __forceinline__ u32x4 tdm_g0(uint64_t gaddr, uint32_t lds_off) {
  u32x4 g;
  g.x = 1u;                                                  // count=1
  g.y = lds_off;                                             // LDS byte address
  g.z = (uint32_t)gaddr;                                     // global_addr[31:0]
  g.w = ((uint32_t)(gaddr >> 32) & 0x1FFFFFFu) | (2u << 30); // [56:32] | type=2
  return g;
}

// Group 1: [17:16]=data_size(2 -> 4B), [79:48]=tensor_dim0, [127:112]=tile_dim0,
//          [207:160]=tensor_dim0_stride. 1-D tile: tile_dim1/2 = 0.
__device__ __forceinline__ i32x8 tdm_g1(uint32_t tile_dw, uint32_t tensor_dw) {
  i32x8 g;
  g[0] = (int)(2u << 16);                     // data_size = 4 bytes
  g[1] = (int)((tensor_dw & 0xFFFFu) << 16);  // tensor_dim0[15:0]  (bits 63:48)
  g[2] = (int)(tensor_dw >> 16);              // tensor_dim0[31:16] (bits 79:64)
  g[3] = (int)((tile_dw & 0xFFFFu) << 16);    // tile_dim0          (bits 127:112)
  g[4] = 0;                                   // tile_dim1 = tile_dim2 = 0
  g[5] = (int)tile_dw;                        // tensor_dim0_stride lo32
  g[6] = 0;
  g[7] = 0;
  return g;
}

__device__ __forceinline__ void tdm_load(u32x4 g0, i32x8 g1) {
  i32x4 z4 = {0, 0, 0, 0};
#if defined(__clang_major__) && (__clang_major__ >= 23)
  i32x8 z8 = {0, 0, 0, 0, 0, 0, 0, 0};
  __builtin_amdgcn_tensor_load_to_lds(g0, g1, z4, z4, z8, CPOL_LOAD_NT);
#else
  __builtin_amdgcn_tensor_load_to_lds(g0, g1, z4, z4, CPOL_LOAD_NT);
#endif
}

__device__ __forceinline__ void tdm_store(u32x4 g0, i32x8 g1) {
  i32x4 z4 = {0, 0, 0, 0};
#if defined(__clang_major__) && (__clang_major__ >= 23)
  i32x8 z8 = {0, 0, 0, 0, 0, 0, 0, 0};
  __builtin_amdgcn_tensor_store_from_lds(g0, g1, z4, z4, z8, CPOL_STORE_NT);
#else
  __builtin_amdgcn_tensor_store_from_lds(g0, g1, z4, z4, CPOL_STORE_NT);
#endif
}

// Numerically stable softplus == logaddexp(x, 0) = max(x,0) + log1p(exp(-|x|)).
// log1p via hw log + correction: e in (0,1], u = 1+e in [1,2] so (u-1) is exact;
// log1p(e) = log(u) + (e - (u-1))/u. Fast rcp is fine: the term is O(2^-24) rel.
// ~11 VALU per call vs ~80 for ocml log1pf; keeps VALU well under DMA time.
__device__ __forceinline__ float softplus_f(float v) {
  const float e = __expf(-fabsf(v));          // v_exp_f32 path
  const float u = 1.0f + e;
  const float l = __logf(u) + (e - (u - 1.0f)) * __builtin_amdgcn_rcpf(u);
  return fmaxf(v, 0.0f) + l;
}

__global__ __launch_bounds__(THREADS) void spd_lltr_kernel(
    const float* __restrict__ x, float* __restrict__ out, unsigned int nSamples) {
  __shared__ float sIn[IN_DW];    // 24 KB
  __shared__ float sOut[OUT_DW];  // 36 KB

  const unsigned int t   = threadIdx.x;
  const unsigned int blk = blockIdx.x;
  const uint64_t inBase   = (uint64_t)blk * IN_DW;   // dword index into x
  const uint64_t outBase  = (uint64_t)blk * OUT_DW;  // dword index into out
  const uint64_t totalIn  = (uint64_t)nSamples * 6u;
  const uint64_t totalOut = (uint64_t)nSamples * 9u;

  // ---- TDM: global -> LDS (issued once per workgroup by wave 0) ----------
  if (t == 0) {
    const uint64_t gaddr  = (uint64_t)(uintptr_t)(x + inBase);
    const uint32_t ldsOff = (uint32_t)(uintptr_t)&sIn[0]; // flat[31:0] = LDS offset
    const uint32_t remain = (uint32_t)(totalIn - inBase); // tail: OOB reads -> 0
    tdm_load(tdm_g0(gaddr, ldsOff), tdm_g1(IN_DW, remain));
  }
  __builtin_amdgcn_s_wait_tensorcnt(0);   // wave 0 waits for its DMA
  __syncthreads();

  // ---- per-sample math; strided mapping keeps LDS bank-conflict free -----
#pragma unroll
  for (int k = 0; k < SPT; ++k) {
    const unsigned int s = t + (unsigned int)k * THREADS;
    const float x0 = sIn[6 * s + 0];
    const float x1 = sIn[6 * s + 1];
    const float x2 = sIn[6 * s + 2];
    const float x5 = sIn[6 * s + 5];

    const float d0 = softplus_f(x0) + 1e-12f;
    const float d2 = softplus_f(x2) + 1e-12f;
    const float d5 = softplus_f(x5) + 1e-12f;

    const float o01 = d0 * x1;
    sOut[9 * s + 0] = d0 * d0;
    sOut[9 * s + 1] = o01;
    sOut[9 * s + 2] = 0.0f;
    sOut[9 * s + 3] = o01;
    sOut[9 * s + 4] = __fmaf_rn(x1, x1, d2 * d2);
    sOut[9 * s + 5] = 0.0f;
    sOut[9 * s + 6] = 0.0f;
    sOut[9 * s + 7] = 0.0f;
    sOut[9 * s + 8] = d5 * d5;
  }

  __syncthreads();

  // ---- TDM: LDS -> global (tail: OOB writes dropped via tensor_dim0) -----
  if (t == 0) {
    const uint64_t gaddr  = (uint64_t)(uintptr_t)(out + outBase);
    const uint32_t ldsOff = (uint32_t)(uintptr_t)&sOut[0];
    const uint32_t remain = (uint32_t)(totalOut - outBase);
    tdm_store(tdm_g0(gaddr, ldsOff), tdm_g1(OUT_DW, remain));
  }
  __builtin_amdgcn_s_wait_tensorcnt(0);   // LDS-reuse safety; ENDPGM waits idle
}

extern "C" void kernel_launch(void* const* d_in, const int* in_sizes, int n_in,
                              void* d_out, int out_size, void* d_ws, size_t ws_size,
                              hipStream_t stream) {
  (void)n_in; (void)out_size; (void)d_ws; (void)ws_size;
  const float* x = (const float*)d_in[0];
  float* out     = (float*)d_out;
  const unsigned int nSamples = (unsigned int)(in_sizes[0] / 6);
  const unsigned int grid     = (nSamples + TILE - 1) / TILE;
  spd_lltr_kernel<<<grid, THREADS, 0, stream>>>(x, out, nSamples);
}